// ParametricLearned_3204045603620
// MI455X (gfx1250) — compile-verified
//
#include <hip/hip_runtime.h>
#include <hip/hip_bf16.h>

// Problem constants (must match the reference)
#define HH     64
#define VV     64
#define INNER  24
#define SEQL   32
#define CAP    8
#define NPAIR  ((SEQL - 2) / 2)                       // 15
#define NPARAM (HH*INNER + INNER + INNER*HH + HH)     // 3160
#define THREADS 256
#define NSLOT  ((NPARAM + THREADS - 1) / THREADS)     // 13

typedef float v2f __attribute__((ext_vector_type(2)));
typedef float v8f __attribute__((ext_vector_type(8)));

// Param layout in sP: [w1: H*INNER][b1: INNER][w2: INNER*H][b2: H]
#define OFF_B1 (HH*INNER)
#define OFF_W2 (HH*INNER + INNER)
#define OFF_B2 (HH*INNER + INNER + INNER*HH)

__global__ __launch_bounds__(THREADS)
void fused_tlearn_kernel(
    const int*   __restrict__ seqs,
    const float* __restrict__ embed,
    const float* __restrict__ ff_w1, const float* __restrict__ ff_b1,
    const float* __restrict__ ff_w2, const float* __restrict__ ff_b2,
    const float* __restrict__ ln_g,  const float* __restrict__ ln_b,
    const float* __restrict__ scorer_w, const float* __restrict__ scorer_b,
    const float* __restrict__ mlp_w1, const float* __restrict__ mlp_b1,
    const float* __restrict__ mlp_w2, const float* __restrict__ mlp_b2,
    const float* __restrict__ out_w,  const float* __restrict__ out_b,
    float* __restrict__ out)
{
    __shared__ int   sTok[SEQL];
    __shared__ float sE[SEQL * HH];         // gathered embeddings  [32][64]
    __shared__ float sF[SEQL * 2 * HH];     // relu(E@w1+b1)        [32][128]
    __shared__ float sH[SEQL * HH];         // encoded tokens       [32][64]
    __shared__ float sP[NPARAM];            // fine-tuned MLP params
    __shared__ float sScore[NPAIR];
    __shared__ int   sBuf[CAP];
    __shared__ float sZ1[INNER];            // pre-relu hidden
    __shared__ float sA1[INNER];            // relu hidden
    __shared__ float sDZ2[HH];
    __shared__ float sDZ1[INNER];
    __shared__ float sQ[HH];                // final mlp output

    const int tid  = threadIdx.x;
    const int lane = tid & 31;
    const int wave = tid >> 5;
    const int s    = blockIdx.x;

    // ---------------- Phase 0: tokens + embedding gather ----------------
    if (tid < SEQL) sTok[tid] = seqs[s * SEQL + tid];
    __syncthreads();
    for (int i = tid; i < SEQL * HH; i += THREADS) {
        int r = i >> 6, c = i & 63;
        sE[i] = embed[sTok[r] * HH + c];
    }
    __syncthreads();

    const int row16 = lane & 15;
    const int hi    = lane >> 4;   // 0 for lanes 0-15, 1 for lanes 16-31

    // ---------------- Phase 1: F = relu(E[32x64] @ ff_w1[64x128] + b1) ----
    // 16 output tiles of 16x16; 8 waves x 2 tiles; K stepped by 4 via
    // V_WMMA_F32_16X16X4_F32.
    for (int r = 0; r < 2; ++r) {
        const int m0 = r * 16;
        const int n0 = wave * 16;
        v8f acc = {};
        for (int k0 = 0; k0 < HH; k0 += 4) {
            const int ka = k0 + hi * 2;
            // A frag: lanes0-15 hold K=k0,k0+1; lanes16-31 hold K=k0+2,k0+3
            v2f af = *(const v2f*)&sE[(m0 + row16) * HH + ka];
            // B frag: VGPR0 = row ka, VGPR1 = row ka+1, col = n0+row16
            const int col = n0 + row16;
            v2f bf;
            bf.x = ff_w1[ka * (2 * HH) + col];
            bf.y = ff_w1[(ka + 1) * (2 * HH) + col];
            acc = __builtin_amdgcn_wmma_f32_16x16x4_f32(
                    false, af, false, bf, (short)0, acc, false, false);
        }
        const int colo = n0 + row16;
        const float bb = ff_b1[colo];
#pragma unroll
        for (int e = 0; e < 8; ++e) {
            const int rowo = m0 + e + hi * 8;
            const float v = acc[e] + bb;
            sF[rowo * (2 * HH) + colo] = v > 0.f ? v : 0.f;
        }
    }
    __syncthreads();

    // ---------------- Phase 2: X = E + F[32x128] @ ff_w2[128x64] + b2 -----
    {
        const int m0 = (wave >> 2) * 16;
        const int n0 = (wave & 3) * 16;
        v8f acc = {};
        for (int k0 = 0; k0 < 2 * HH; k0 += 4) {
            const int ka = k0 + hi * 2;
            v2f af = *(const v2f*)&sF[(m0 + row16) * (2 * HH) + ka];
            const int col = n0 + row16;
            v2f bf;
            bf.x = ff_w2[ka * HH + col];
            bf.y = ff_w2[(ka + 1) * HH + col];
            acc = __builtin_amdgcn_wmma_f32_16x16x4_f32(
                    false, af, false, bf, (short)0, acc, false, false);
        }
        const int colo = n0 + row16;
        const float bb = ff_b2[colo];
#pragma unroll
        for (int e = 0; e < 8; ++e) {
            const int rowo = m0 + e + hi * 8;
            sH[rowo * HH + colo] = sE[rowo * HH + colo] + acc[e] + bb;
        }
    }
    __syncthreads();

    // ---------------- Phase 3: LayerNorm per token row --------------------
    if (tid < SEQL) {
        float mu = 0.f;
        for (int c = 0; c < HH; ++c) mu += sH[tid * HH + c];
        mu *= (1.f / HH);
        float var = 0.f;
        for (int c = 0; c < HH; ++c) {
            const float d = sH[tid * HH + c] - mu;
            var += d * d;
        }
        var *= (1.f / HH);
        const float rs = rsqrtf(var + 1e-5f);
        for (int c = 0; c < HH; ++c)
            sH[tid * HH + c] = (sH[tid * HH + c] - mu) * rs * ln_g[c] + ln_b[c];
    }
    __syncthreads();

    // ---------------- Phase 4: pair scores --------------------------------
    if (tid < NPAIR) {
        const float* kr = &sH[(2 * tid) * HH];
        const float* vr = &sH[(2 * tid + 1) * HH];
        float sc = scorer_b[0];
        for (int c = 0; c < HH; ++c)
            sc += kr[c] * scorer_w[c] + vr[c] * scorer_w[HH + c];
        sScore[tid] = sc;
    }
    __syncthreads();

    // ---------------- Phase 5: evict-min buffer scan (thread 0) -----------
    if (tid == 0) {
        int   bi[CAP];
        float bs[CAP];
        for (int i = 0; i < CAP; ++i) { bi[i] = i; bs[i] = sScore[i]; }
        for (int n = CAP; n < NPAIR; ++n) {
            int mi = 0;
            for (int i = 1; i < CAP; ++i) if (bs[i] < bs[mi]) mi = i;  // first min
            for (int i = mi; i < CAP - 1; ++i) { bi[i] = bi[i + 1]; bs[i] = bs[i + 1]; }
            bi[CAP - 1] = n; bs[CAP - 1] = sScore[n];
        }
        for (int i = 0; i < CAP; ++i) sBuf[i] = bi[i];
    }

    // ---------------- Phase 6: init fine-tune params ----------------------
    for (int q = tid; q < NPARAM; q += THREADS) {
        float pv;
        if      (q < OFF_B1) pv = mlp_w1[q];
        else if (q < OFF_W2) pv = mlp_b1[q - OFF_B1];
        else if (q < OFF_B2) pv = mlp_w2[q - OFF_W2];
        else                 pv = mlp_b2[q - OFF_B2];
        sP[q] = pv;
    }
    __syncthreads();

    // ---------------- Phase 7: 8 Adam steps -------------------------------
    float am[NSLOT], av[NSLOT];
#pragma unroll
    for (int r = 0; r < NSLOT; ++r) { am[r] = 0.f; av[r] = 0.f; }

    float bp1 = 1.f, bp2 = 1.f;
    for (int step = 0; step < CAP; ++step) {
        const int pi = sBuf[step];
        const float* kv = &sH[(2 * pi) * HH];
        const float* tv = &sH[(2 * pi + 1) * HH];

        // forward hidden: z1 = k @ w1 + b1 ; a1 = relu(z1)
        if (tid < INNER) {
            float z = sP[OFF_B1 + tid];
            for (int i = 0; i < HH; ++i) z += kv[i] * sP[i * INNER + tid];
            sZ1[tid] = z;
            sA1[tid] = z > 0.f ? z : 0.f;
        }
        __syncthreads();
        // output + dz2 = (2/H) * (z2 - v)
        if (tid < HH) {
            float z = sP[OFF_B2 + tid];
            for (int i = 0; i < INNER; ++i) z += sA1[i] * sP[OFF_W2 + i * HH + tid];
            sDZ2[tid] = (2.f / HH) * (z - tv[tid]);
        }
        __syncthreads();
        // dz1 = (w2 @ dz2) * relu'(z1)
        if (tid < INNER) {
            float d = 0.f;
            for (int j = 0; j < HH; ++j) d += sP[OFF_W2 + tid * HH + j] * sDZ2[j];
            sDZ1[tid] = sZ1[tid] > 0.f ? d : 0.f;
        }
        __syncthreads();

        // Adam update; each thread owns NSLOT param slots; m/v live in VGPRs
        bp1 *= 0.9f;
        bp2 *= 0.999f;
        const float c1 = 1.f / (1.f - bp1);
        const float c2 = 1.f / (1.f - bp2);
#pragma unroll
        for (int r = 0; r < NSLOT; ++r) {
            const int q = tid + r * THREADS;
            if (q < NPARAM) {
                float g;
                if (q < OFF_B1) {                       // w1: k[i] * dz1[j]
                    const int i = q / INNER, j = q - i * INNER;
                    g = kv[i] * sDZ1[j];
                } else if (q < OFF_W2) {                // b1: dz1
                    g = sDZ1[q - OFF_B1];
                } else if (q < OFF_B2) {                // w2: a1[i] * dz2[j]
                    const int qq = q - OFF_W2;
                    const int i = qq >> 6, j = qq & 63;
                    g = sA1[i] * sDZ2[j];
                } else {                                // b2: dz2
                    g = sDZ2[q - OFF_B2];
                }
                const float m_ = 0.9f * am[r] + 0.1f * g;
                const float v_ = 0.999f * av[r] + 0.001f * g * g;
                am[r] = m_;
                av[r] = v_;
                sP[q] -= 0.05f * (m_ * c1) / (sqrtf(v_ * c2) + 1e-8f);
            }
        }
        __syncthreads();
    }

    // ---------------- Phase 8: query forward + output head ----------------
    {
        const float* kq = &sH[(SEQL - 2) * HH];
        if (tid < INNER) {
            float z = sP[OFF_B1 + tid];
            for (int i = 0; i < HH; ++i) z += kq[i] * sP[i * INNER + tid];
            sA1[tid] = z > 0.f ? z : 0.f;
        }
        __syncthreads();
        if (tid < HH) {
            float z = sP[OFF_B2 + tid];
            for (int i = 0; i < INNER; ++i) z += sA1[i] * sP[OFF_W2 + i * HH + tid];
            sQ[tid] = z;
        }
        __syncthreads();
        if (tid < VV) {
            float o = out_b[tid];
            for (int i = 0; i < HH; ++i) o += sQ[i] * out_w[i * VV + tid];
            out[s * VV + tid] = o;
        }
    }
}

extern "C" void kernel_launch(void* const* d_in, const int* in_sizes, int n_in,
                              void* d_out, int out_size, void* d_ws, size_t ws_size,
                              hipStream_t stream) {
    const int*   seqs     = (const int*)  d_in[0];
    const float* embed    = (const float*)d_in[1];
    const float* ff_w1    = (const float*)d_in[2];
    const float* ff_b1    = (const float*)d_in[3];
    const float* ff_w2    = (const float*)d_in[4];
    const float* ff_b2    = (const float*)d_in[5];
    const float* ln_g     = (const float*)d_in[6];
    const float* ln_b     = (const float*)d_in[7];
    const float* scorer_w = (const float*)d_in[8];
    const float* scorer_b = (const float*)d_in[9];
    const float* mlp_w1   = (const float*)d_in[10];
    const float* mlp_b1   = (const float*)d_in[11];
    const float* mlp_w2   = (const float*)d_in[12];
    const float* mlp_b2   = (const float*)d_in[13];
    const float* out_w    = (const float*)d_in[14];
    const float* out_b    = (const float*)d_in[15];
    float*       out      = (float*)d_out;

    const int B = in_sizes[0] / SEQL;   // 16384 samples

    fused_tlearn_kernel<<<B, THREADS, 0, stream>>>(
        seqs, embed, ff_w1, ff_b1, ff_w2, ff_b2, ln_g, ln_b,
        scorer_w, scorer_b, mlp_w1, mlp_b1, mlp_w2, mlp_b2,
        out_w, out_b, out);
}